// RanTSelecor_37529424232932
// MI455X (gfx1250) — compile-verified
//
#include <hip/hip_runtime.h>

// ---------------------------------------------------------------------------
// Problem constants (fixed by setup_inputs): F=256 frames, T=576 tokens,
// D=1408 features, K=144 selected tokens. Output layout [8,32,144,1408] is
// flat-identical to [F,K,D].
// Roofline: traffic-dominated (~830MB stream + ~208MB gather-read + ~208MB
// write ≈ 53us at 23.3TB/s); GEMM/GEMV/topk compute is negligible. Softmax is
// skipped (monotonic -> same top-k). The logits GEMM runs in full fp32 on
// V_WMMA_F32_16X16X4_F32 so selection semantics match the fp32 reference.
// ---------------------------------------------------------------------------
#define F_TOTAL 256
#define T_TOK   576
#define D_FEAT  1408
#define K_SEL   144

typedef __attribute__((ext_vector_type(2))) float v2f;
typedef __attribute__((ext_vector_type(8))) float v8f;

// ---------------------------------------------------------------------------
// Kernel 1: s[f,t] = dot(embeds[f,t,:], pre_w) + pre_b
// Bandwidth-bound (830 MB read). One wave per (f,t) row: 11 float4 loads/lane,
// then a wave32 shuffle reduction.
// ---------------------------------------------------------------------------
__global__ __launch_bounds__(256) void k1_preproj(const float* __restrict__ embeds,
                                                  const float* __restrict__ pre_w,
                                                  const float* __restrict__ pre_b,
                                                  float* __restrict__ s)
{
    const int wave = threadIdx.x >> 5;
    const int lane = threadIdx.x & 31;
    const int row  = blockIdx.x * 8 + wave;             // < 147456
    const float4* rp = (const float4*)(embeds + (long)row * D_FEAT);
    const float4* wp = (const float4*)pre_w;
    float acc = 0.f;
#pragma unroll
    for (int i = 0; i < D_FEAT / 4 / 32; ++i) {         // 11 iterations
        float4 e = rp[lane + i * 32];
        float4 w = wp[lane + i * 32];
        acc += e.x * w.x + e.y * w.y + e.z * w.z + e.w * w.w;
    }
#pragma unroll
    for (int off = 16; off >= 1; off >>= 1)
        acc += __shfl_xor(acc, off, 32);
    if (lane == 0) s[row] = acc + pre_b[0];
}

// ---------------------------------------------------------------------------
// Kernel 2a: logits[F,T] = s[F,T] @ W[T,T] + bias, fp32 WMMA.
// Block = 16 frames x all 576 columns, 128 threads = 4 waves.
// 36 N-tiles split EXACTLY 9 per wave -> no guards anywhere in the WMMA loop:
// all loops fully unroll with constant accumulator indices, EXEC is trivially
// all-1s, and the scheduler can batch the B-operand ds_loads ahead of the
// WMMA chain (previous version drowned in per-WMMA scalar branches).
// K loop in chunks of 8 staged in LDS; A (s tile) staged once.
// LDS: sA 16x577 f32 (36.9KB) + sB 8x592 f32 (18.9KB) = 55.8KB.
// ---------------------------------------------------------------------------
__global__ __launch_bounds__(128) void k2a_gemm(const float* __restrict__ s,
                                                const float* __restrict__ W,
                                                const float* __restrict__ bias,
                                                float* __restrict__ logits)
{
    __shared__ float sA[16 * 577];
    __shared__ __align__(16) float sB[8 * 592];

    const int tid  = threadIdx.x;     // 0..127
    const int wave = tid >> 5;        // 0..3
    const int lane = tid & 31;
    const int half = lane >> 4;       // K/M split within the wave
    const int lm   = lane & 15;
    const int fbase = blockIdx.x * 16;

    // Stage A: s rows [fbase .. fbase+15], padded stride 577 (odd -> clean banks)
    for (int i = tid; i < 16 * T_TOK; i += 128) {
        int r = i / T_TOK, c = i - r * T_TOK;
        sA[r * 577 + c] = s[(fbase + r) * T_TOK + c];
    }

    v8f zero = {};
    v8f acc[9];
#pragma unroll
    for (int t = 0; t < 9; ++t) acc[t] = zero;

    for (int k0 = 0; k0 < T_TOK; k0 += 8) {
        __syncthreads();
        // Stage B rows k0..k0+7 (contiguous rows of W) as float4; padded row
        // stride 592 floats keeps the two half-wave reads on disjoint banks.
        for (int i = tid; i < 8 * (T_TOK / 4); i += 128) {
            int r = i / (T_TOK / 4), c = i - r * (T_TOK / 4);
            ((float4*)&sB[r * 592])[c] = ((const float4*)(W + (k0 + r) * T_TOK))[c];
        }
        __syncthreads();
#pragma unroll
        for (int kk = 0; kk < 8; kk += 4) {
            // A 16x4 f32 layout: lanes 0-15 -> K=0/1 in v[0]/v[1]; lanes 16-31 -> K=2/3
            v2f a;
            a.x = sA[lm * 577 + k0 + kk + 2 * half];
            a.y = sA[lm * 577 + k0 + kk + 2 * half + 1];
#pragma unroll
            for (int t = 0; t < 9; ++t) {
                const int n0 = (wave * 9 + t) * 16;
                // B 4x16 f32 layout mirrors A with M<->N.
                v2f b;
                b.x = sB[(kk + 2 * half) * 592 + n0 + lm];
                b.y = sB[(kk + 2 * half + 1) * 592 + n0 + lm];
                acc[t] = __builtin_amdgcn_wmma_f32_16x16x4_f32(
                    /*neg_a=*/false, a, /*neg_b=*/false, b,
                    /*c_mod=*/(short)0, acc[t],
                    /*reuse_a=*/false, /*reuse_b=*/false);
            }
        }
    }

    // C/D layout: lanes 0-15 VGPR r -> [M=r][N=lane]; lanes 16-31 -> [M=8+r][N=lane-16]
#pragma unroll
    for (int t = 0; t < 9; ++t) {
        const int col = (wave * 9 + t) * 16 + lm;
        const float bcol = bias[col];
#pragma unroll
        for (int r = 0; r < 8; ++r) {
            const int row = fbase + r + 8 * half;
            logits[row * T_TOK + col] = acc[t][r] + bcol;
        }
    }
}

// ---------------------------------------------------------------------------
// Kernel 2b: per-frame top-K selection. Softmax skipped (monotonic).
// rank[i] = #{j : lv[j] > lv[i] or (lv[j]==lv[i] and j<i)}  (stable-descending
// order statistic, matching jnp.argsort(-scores)). Ranks are a permutation so
// exactly K entries have rank < K; ascending-index output position = prefix
// count of selection flags. One 576-thread block per frame; scans hit LDS.
// ---------------------------------------------------------------------------
__global__ __launch_bounds__(576) void k2b_topk(const float* __restrict__ logits,
                                                int* __restrict__ idx)
{
    __shared__ float lv[T_TOK];
    __shared__ unsigned char selp[T_TOK];
    const int f = blockIdx.x;
    const int i = threadIdx.x;
    float v = logits[f * T_TOK + i];
    lv[i] = v;
    __syncthreads();
    int rank = 0;
    for (int j = 0; j < T_TOK; ++j) {
        float o = lv[j];
        rank += (o > v) || ((o == v) && (j < i));
    }
    int sel = (rank < K_SEL) ? 1 : 0;
    selp[i] = (unsigned char)sel;
    __syncthreads();
    if (sel) {
        int pos = 0;
        for (int j = 0; j < i; ++j) pos += selp[j];
        idx[f * K_SEL + pos] = i;
    }
}

// ---------------------------------------------------------------------------
// Kernel 3: gather selected rows. One wave per output row (256*144 rows),
// 11 float4 loads + stores per lane. ~208 MB read + 208 MB write.
// ---------------------------------------------------------------------------
__global__ __launch_bounds__(256) void k3_gather(const float* __restrict__ embeds,
                                                 const int* __restrict__ idx,
                                                 float* __restrict__ out)
{
    const int wave = threadIdx.x >> 5;
    const int lane = threadIdx.x & 31;
    const int r = blockIdx.x * 8 + wave;                // < 36864
    const int f = r / K_SEL;
    const int tok = idx[r];
    const float4* src = (const float4*)(embeds + ((long)f * T_TOK + tok) * D_FEAT);
    float4* dst = (float4*)(out + (long)r * D_FEAT);
#pragma unroll
    for (int i = 0; i < D_FEAT / 4 / 32; ++i)
        dst[lane + i * 32] = src[lane + i * 32];
}

// ---------------------------------------------------------------------------
// Launch. Workspace layout (needs ~1.33 MB):
//   [0, 576KB)      s       [F,T] f32
//   [576KB, 1152KB) logits  [F,T] f32
//   [1152KB, ...)   idx     [F,K] i32
// ---------------------------------------------------------------------------
extern "C" void kernel_launch(void* const* d_in, const int* in_sizes, int n_in,
                              void* d_out, int out_size, void* d_ws, size_t ws_size,
                              hipStream_t stream)
{
    (void)in_sizes; (void)n_in; (void)out_size; (void)ws_size;
    const float* embeds  = (const float*)d_in[0];
    const float* pre_w   = (const float*)d_in[1];
    const float* pre_b   = (const float*)d_in[2];
    const float* trans_w = (const float*)d_in[3];
    const float* trans_b = (const float*)d_in[4];
    // d_in[5] (n_frames) and d_in[6] (out_token_num) are shape params already
    // baked into the constants; the final reshape doesn't change flat layout.

    char* ws = (char*)d_ws;
    float* s      = (float*)(ws);
    float* logits = (float*)(ws + (size_t)F_TOTAL * T_TOK * 4);
    int*   idx    = (int*)  (ws + (size_t)F_TOTAL * T_TOK * 8);
    float* out    = (float*)d_out;

    k1_preproj<<<F_TOTAL * T_TOK / 8, 256, 0, stream>>>(embeds, pre_w, pre_b, s);
    k2a_gemm  <<<F_TOTAL / 16,        128, 0, stream>>>(s, trans_w, trans_b, logits);
    k2b_topk  <<<F_TOTAL,           T_TOK, 0, stream>>>(logits, idx);
    k3_gather <<<F_TOTAL * K_SEL / 8, 256, 0, stream>>>(embeds, idx, out);
}